// GlobalMelSpecDiscretizer_16252156248526
// MI455X (gfx1250) — compile-verified
//
#include <hip/hip_runtime.h>
#include <hip/hip_bf16.h>
#include <stdint.h>

#define KPAD    64
#define THREADS 256

// ---------------- CDNA5 async global->LDS (ASYNCcnt) helpers ----------------
typedef int v4i __attribute__((ext_vector_type(4)));
typedef __attribute__((address_space(1))) v4i*  as1_v4i_ptr;
typedef __attribute__((address_space(3))) v4i*  as3_v4i_ptr;
typedef __attribute__((address_space(3))) void* as3_void_ptr;

#if defined(__gfx1250__) && __has_builtin(__builtin_amdgcn_global_load_async_to_lds_b128)
#define ASYNC_BUILTIN 1
#else
#define ASYNC_BUILTIN 0
#endif

__device__ __forceinline__ void async_copy_b128(const void* gsrc, void* ldst) {
#if ASYNC_BUILTIN
  // signature (probe-confirmed by diagnostic): (v4i AS1*, v4i AS3*, imm offset, imm cpol)
  __builtin_amdgcn_global_load_async_to_lds_b128(
      (as1_v4i_ptr)(uintptr_t)gsrc,
      (as3_v4i_ptr)(as3_void_ptr)ldst, 0, 0);
#else
  uint32_t lds_off = (uint32_t)(uintptr_t)(as3_void_ptr)ldst;
  asm volatile("global_load_async_to_lds_b128 %0, %1, off"
               :
               : "v"(lds_off), "v"(gsrc)
               : "memory");
#endif
}

__device__ __forceinline__ void wait_async_le0() {
#if defined(__gfx1250__) && __has_builtin(__builtin_amdgcn_s_wait_asynccnt)
  __builtin_amdgcn_s_wait_asynccnt(0);
#else
  asm volatile("s_wait_asynccnt 0" ::: "memory");
#endif
}

__device__ __forceinline__ void wait_async_le1() {
#if defined(__gfx1250__) && __has_builtin(__builtin_amdgcn_s_wait_asynccnt)
  __builtin_amdgcn_s_wait_asynccnt(1);
#else
  asm volatile("s_wait_asynccnt 1" ::: "memory");
#endif
}

// Branchless uniform binary search: #(mids < x) over a 64-entry sorted table
// (real boundaries padded with +INF). 6 dependent ds_load_b32 per element.
__device__ __forceinline__ float lookup(float x,
                                        const float* __restrict__ s_mid,
                                        const float* __restrict__ s_sorted) {
  int base = 0;
#pragma unroll
  for (int s = 32; s > 0; s >>= 1) {
    base += (s_mid[base + s - 1] < x) ? s : 0;
  }
  return s_sorted[base];
}

__global__ __launch_bounds__(THREADS) void discretize_kernel(
    const float4* __restrict__ mel4,
    const float* __restrict__ cen,
    float4* __restrict__ out4,
    int K, int n4, int totalTiles) {
  __shared__ float  s_sorted[KPAD];
  __shared__ float  s_mid[KPAD];
  __shared__ float4 s_buf[2][THREADS];

  const int t = threadIdx.x;
  const float INF = __builtin_inff();

  // ---- build sorted centroid table + midpoints in LDS (rank sort, K<=64) ----
  float* raw = (float*)&s_buf[0][0];  // reuse tile buffer as scratch pre-loop
  if (t < KPAD) raw[t] = (t < K) ? cen[t] : INF;
  __syncthreads();
  if (t < KPAD) {
    float v = raw[t];
    int rank = 0;
#pragma unroll 8
    for (int j = 0; j < KPAD; ++j) {
      float cj = raw[j];
      rank += (cj < v || (cj == v && j < t)) ? 1 : 0;
    }
    s_sorted[rank] = v;  // ranks are a permutation (ties broken by index)
  }
  __syncthreads();
  if (t < KPAD) {
    float lo = s_sorted[t];
    float hi = (t < KPAD - 1) ? s_sorted[t + 1] : INF;
    s_mid[t] = 0.5f * (lo + hi);  // padded slots become +INF
  }
  __syncthreads();

  // ---- grid-stride over tiles, double-buffered async global->LDS ----
  int tile = blockIdx.x;
  const int stride = gridDim.x;
  if (tile >= totalTiles) return;

  {  // prologue: first tile of this block into buffer 0
    int idx = tile * THREADS + t;
    if (idx >= n4) idx = n4 - 1;  // clamp partial tail tile
    async_copy_b128(mel4 + idx, &s_buf[0][t]);
  }

  int buf = 0;
  for (; tile < totalTiles; tile += stride, buf ^= 1) {
    const int nextTile = tile + stride;
    if (nextTile < totalTiles) {
      int idx = nextTile * THREADS + t;
      if (idx >= n4) idx = n4 - 1;
      async_copy_b128(mel4 + idx, &s_buf[buf ^ 1][t]);
      wait_async_le1();  // oldest (current tile) complete; next stays in flight
    } else {
      wait_async_le0();
    }
    asm volatile("" ::: "memory");

    // each lane consumes exactly the 16 bytes its own async copy deposited
    float4 x = s_buf[buf][t];
    float4 r;
    r.x = lookup(x.x, s_mid, s_sorted);
    r.y = lookup(x.y, s_mid, s_sorted);
    r.z = lookup(x.z, s_mid, s_sorted);
    r.w = lookup(x.w, s_mid, s_sorted);

    int idx = tile * THREADS + t;
    if (idx < n4) out4[idx] = r;
  }
}

extern "C" void kernel_launch(void* const* d_in, const int* in_sizes, int n_in,
                              void* d_out, int out_size, void* d_ws, size_t ws_size,
                              hipStream_t stream) {
  (void)n_in; (void)out_size; (void)d_ws; (void)ws_size;

  const float* mel = (const float*)d_in[0];   // [B,M,T] float32
  const float* cen = (const float*)d_in[1];   // [K] float32
  float4* out4 = (float4*)d_out;              // [B,M,T] float32

  int n = in_sizes[0];            // 2,621,440 (divisible by 4: T=2048)
  int K = in_sizes[1];            // 64
  if (K > KPAD) K = KPAD;

  int n4 = n >> 2;
  int totalTiles = (n4 + THREADS - 1) / THREADS;     // 2560
  int grid = (totalTiles + 3) / 4;                   // ~4 tiles/block -> pipelining
  if (grid < 1) grid = 1;

  discretize_kernel<<<grid, THREADS, 0, stream>>>(
      (const float4*)mel, cen, out4, K, n4, totalTiles);
}